// CHMLearner_32031866094250
// MI455X (gfx1250) — compile-verified
//
#include <hip/hip_runtime.h>
#include <hip/hip_bf16.h>
#include <math.h>

typedef float v2f __attribute__((ext_vector_type(2)));
typedef float v8f __attribute__((ext_vector_type(8)));

#define WMMA_F32(a, b, c) \
  __builtin_amdgcn_wmma_f32_16x16x4_f32(false, (a), false, (b), (short)0, (c), false, false)

// ---------------------------------------------------------------------------
// TDM helper: DMA a 16-row x tile_k-col f32 tile (row stride = stride_elems)
// from global memory into LDS at byte offset lds_off, via the Tensor Data
// Mover (D# packed per CDNA5 ISA).
// Falls back to nothing where the builtin doesn't exist (caller handles copy).
// ---------------------------------------------------------------------------
#if __has_builtin(__builtin_amdgcn_tensor_load_to_lds)
#define HAVE_TDM 1
#else
#define HAVE_TDM 0
#endif

__device__ __forceinline__ void tdm_load_tile16(unsigned lds_off,
                                                unsigned long long gaddr,
                                                unsigned tile_k,
                                                unsigned stride_elems)
{
#if HAVE_TDM
  typedef unsigned int u32x4 __attribute__((ext_vector_type(4)));
  typedef int i32x4 __attribute__((ext_vector_type(4)));
  typedef int i32x8 __attribute__((ext_vector_type(8)));
  u32x4 g0;
  g0.x = 1u;                                            // count=1 (valid user D#)
  g0.y = lds_off;                                       // lds_addr
  g0.z = (unsigned)(gaddr & 0xffffffffull);             // global_addr[31:0]
  g0.w = (unsigned)((gaddr >> 32) & 0x1ffffffull)       // global_addr[56:32]
       | (2u << 30);                                    // type = 2 ("image")
  i32x8 g1;
  g1[0] = (int)(2u << 16);                              // wg_mask=0, data_size=4B
  g1[1] = (int)((stride_elems & 0xffffu) << 16);        // tensor_dim0[15:0]
  g1[2] = (int)((stride_elems >> 16) | (16u << 16));    // tensor_dim0 hi | tensor_dim1=16
  g1[3] = (int)(tile_k << 16);                          // tile_dim0 = tile_k
  g1[4] = 16;                                           // tile_dim1 = 16, tile_dim2 = 0
  g1[5] = (int)stride_elems;                            // tensor_dim0_stride lo
  g1[6] = 0;                                            // stride hi | dim1_stride lo
  g1[7] = 0;
  i32x4 z4 = { 0, 0, 0, 0 };
#if __clang_major__ >= 23
  i32x8 z8 = { 0, 0, 0, 0, 0, 0, 0, 0 };
  __builtin_amdgcn_tensor_load_to_lds(g0, g1, z4, z4, z8, 0);
#else
  __builtin_amdgcn_tensor_load_to_lds(g0, g1, z4, z4, 0);
#endif
#else
  (void)lds_off; (void)gaddr; (void)tile_k; (void)stride_elems;
#endif
}

__device__ __forceinline__ void tdm_wait()
{
#if HAVE_TDM
  __builtin_amdgcn_s_wait_tensorcnt(0);
#endif
}

// ---------------------------------------------------------------------------
// Generic strided bilinear (align_corners=True) resampler:
//   out[b, c, oy, ox] = bilerp over in[b*sb + c*sc + y*sy + x*sx], HxW -> OHxOW
// Serves: feature pyramid resize, both interpolate4d stages (via stride tricks).
// ---------------------------------------------------------------------------
__global__ void bilerp_kernel(const float* __restrict__ in, float* __restrict__ out,
                              int B, int CH, int H, int Wd, int OH, int OW,
                              long sb, long sc, long sy, long sx, long ob, int clip)
{
  long idx = (long)blockIdx.x * blockDim.x + threadIdx.x;
  long total = (long)B * CH * OH * OW;
  if (idx >= total) return;
  int ox = (int)(idx % OW); long t = idx / OW;
  int oy = (int)(t % OH);   t /= OH;
  int c  = (int)(t % CH);   int b = (int)(t / CH);
  float fy = (OH > 1) ? (float)oy * (float)(H - 1) / (float)(OH - 1) : 0.f;
  float fx = (OW > 1) ? (float)ox * (float)(Wd - 1) / (float)(OW - 1) : 0.f;
  int y0 = (int)floorf(fy); y0 = y0 < 0 ? 0 : (y0 > H - 1 ? H - 1 : y0);
  int x0 = (int)floorf(fx); x0 = x0 < 0 ? 0 : (x0 > Wd - 1 ? Wd - 1 : x0);
  int y1 = y0 + 1 < H ? y0 + 1 : H - 1;
  int x1 = x0 + 1 < Wd ? x0 + 1 : Wd - 1;
  float wy = fy - (float)y0, wx = fx - (float)x0;
  const float* p = in + (long)b * sb + (long)c * sc;
  float v00 = p[(long)y0 * sy + (long)x0 * sx];
  float v01 = p[(long)y0 * sy + (long)x1 * sx];
  float v10 = p[(long)y1 * sy + (long)x0 * sx];
  float v11 = p[(long)y1 * sy + (long)x1 * sx];
  float r = (v00 * (1.f - wx) + v01 * wx) * (1.f - wy)
          + (v10 * (1.f - wx) + v11 * wx) * wy;
  if (clip) r = fmaxf(r, 0.f);
  out[(long)b * ob + ((long)c * OH + oy) * OW + ox] = r;
}

// ---------------------------------------------------------------------------
// 3x3 conv 512->128 as 9-tap WMMA GEMM: D[16oc x 16px] tile per wave.
// x: [B,512,Np], w: [128,512,3,3], y: [B,128,Np]. grid (8, ceil(Np/16), B), block 32.
// ---------------------------------------------------------------------------
__global__ void conv2d_wmma_kernel(const float* __restrict__ x, const float* __restrict__ w,
                                   float* __restrict__ y, int sz, int Np)
{
  const int tm = blockIdx.x, tn = blockIdx.y, b = blockIdx.z;
  const int lane = threadIdx.x;
  const int r  = lane & 15;
  const int kh = (lane >> 4) << 1;        // K offset within group of 4
  const int oc = tm * 16 + r;             // A-row (M)
  const int p  = tn * 16 + r;             // B-col (N) = pixel
  const int pc = p < Np ? p : Np - 1;
  const int py = pc / sz, px = pc % sz;
  const bool pv = p < Np;
  const float* wb = w + (long)oc * 512 * 9;
  const float* xb = x + (long)b * 512 * Np;
  v8f acc = {};
  for (int ky = 0; ky < 3; ++ky) {
    for (int kx = 0; kx < 3; ++kx) {
      int iy = py + ky - 1, ix = px + kx - 1;
      bool inb = pv && iy >= 0 && iy < sz && ix >= 0 && ix < sz;
      int iyc = iy < 0 ? 0 : (iy >= sz ? sz - 1 : iy);
      int ixc = ix < 0 ? 0 : (ix >= sz ? sz - 1 : ix);
      const float* wt = wb + ky * 3 + kx;
      const float* xt = xb + iyc * sz + ixc;
      for (int k0 = 0; k0 < 512; k0 += 4) {
        int c0 = k0 + kh;
        if ((k0 & 127) == 0) {
          __builtin_prefetch(xt + (long)(c0 + 128) * Np, 0, 0);
          __builtin_prefetch(wt + (long)(c0 + 128) * 9, 0, 0);
        }
        v2f a, bv;
        a.x = wt[(long)c0 * 9];
        a.y = wt[(long)(c0 + 1) * 9];
        float b0 = xt[(long)c0 * Np];
        float b1 = xt[(long)(c0 + 1) * Np];
        bv.x = inb ? b0 : 0.f;
        bv.y = inb ? b1 : 0.f;
        acc = WMMA_F32(a, bv, acc);
      }
    }
  }
  const int n = tn * 16 + r;
  if (n < Np) {
    const int mb = tm * 16 + ((lane >> 4) << 3);
    for (int j = 0; j < 8; ++j)
      y[((long)b * 128 + (mb + j)) * Np + n] = acc[j];
  }
}

// L2 norms over 128 channels: f [B,128,Np] -> nrm [B,Np]
__global__ void norm_kernel(const float* __restrict__ f, float* __restrict__ nrm, int Np)
{
  int idx = blockIdx.x * blockDim.x + threadIdx.x;
  if (idx >= 4 * Np) return;
  int b = idx / Np, p = idx % Np;
  const float* fp = f + (long)b * 128 * Np + p;
  float s = 0.f;
  for (int c = 0; c < 128; ++c) { float v = fp[(long)c * Np]; s += v * v; }
  nrm[idx] = sqrtf(s);
}

// ---------------------------------------------------------------------------
// Correlation: corr[b,n,m] = <s[:,n], t[:,m]> / (|s_n||t_m|), K=128 via WMMA f32.
// sf: [B,128,Ns], tf: [B,128,Nt]. grid (ceil(Ns/16), ceil(Nt/16), B), block 32.
// ---------------------------------------------------------------------------
__global__ void corr_wmma_kernel(const float* __restrict__ sf, const float* __restrict__ tf,
                                 const float* __restrict__ sn, const float* __restrict__ tn,
                                 float* __restrict__ corr, int Ns, int Nt)
{
  const int tmb = blockIdx.x, tnb = blockIdx.y, b = blockIdx.z;
  const int lane = threadIdx.x;
  const int r  = lane & 15;
  const int kh = (lane >> 4) << 1;
  int n = tmb * 16 + r; bool nv = n < Ns; int ncl = nv ? n : Ns - 1;
  int m = tnb * 16 + r; bool mv = m < Nt; int mcl = mv ? m : Nt - 1;
  const float* sp = sf + (long)b * 128 * Ns + ncl;
  const float* tp = tf + (long)b * 128 * Nt + mcl;
  v8f acc = {};
  for (int k0 = 0; k0 < 128; k0 += 4) {
    int c0 = k0 + kh;
    v2f a, bv;
    float a0 = sp[(long)c0 * Ns], a1 = sp[(long)(c0 + 1) * Ns];
    float b0 = tp[(long)c0 * Nt], b1 = tp[(long)(c0 + 1) * Nt];
    a.x  = nv ? a0 : 0.f; a.y  = nv ? a1 : 0.f;
    bv.x = mv ? b0 : 0.f; bv.y = mv ? b1 : 0.f;
    acc = WMMA_F32(a, bv, acc);
  }
  int col = tnb * 16 + r;
  if (col < Nt) {
    float tnv = tn[(long)b * Nt + col];
    for (int j = 0; j < 8; ++j) {
      int row = tmb * 16 + j + ((lane >> 4) << 3);
      if (row < Ns) {
        float snv = sn[(long)b * Ns + row];
        corr[((long)b * Ns + row) * Nt + col] = acc[j] / (snv * tnv);
      }
    }
  }
}

// ---------------------------------------------------------------------------
// 6D conv ('same', kernel [3,3,5,5,5,5] in LDS) + bias + sigmoid.
// in/out: [B,3,3,16,16,16,16]
// ---------------------------------------------------------------------------
__global__ void conv6d_kernel(const float* __restrict__ in, const float* __restrict__ k6,
                              const float* __restrict__ bias, float* __restrict__ out)
{
  __shared__ float kk[5625];
  for (int t = threadIdx.x; t < 5625; t += blockDim.x) kk[t] = k6[t];
  __syncthreads();
  long idx = (long)blockIdx.x * blockDim.x + threadIdx.x;
  if (idx >= (long)4 * 9 * 65536) return;
  int sp = (int)(idx & 65535);
  int rest = (int)(idx >> 16);
  int a2 = rest % 3; rest /= 3;
  int a1 = rest % 3; int b = rest / 3;
  int l = sp & 15, k = (sp >> 4) & 15, j = (sp >> 8) & 15, i = (sp >> 12) & 15;
  int dalo = 1 - a1 > 0 ? 1 - a1 : 0, dahi = 4 - a1 < 3 ? 4 - a1 : 3;
  int dblo = 1 - a2 > 0 ? 1 - a2 : 0, dbhi = 4 - a2 < 3 ? 4 - a2 : 3;
  int d1lo = 2 - i > 0 ? 2 - i : 0, d1hi = 18 - i < 5 ? 18 - i : 5;
  int d2lo = 2 - j > 0 ? 2 - j : 0, d2hi = 18 - j < 5 ? 18 - j : 5;
  int d3lo = 2 - k > 0 ? 2 - k : 0, d3hi = 18 - k < 5 ? 18 - k : 5;
  int d4lo = 2 - l > 0 ? 2 - l : 0, d4hi = 18 - l < 5 ? 18 - l : 5;
  float sum = 0.f;
  for (int da = dalo; da < dahi; ++da)
    for (int db = dblo; db < dbhi; ++db) {
      const float* ib = in + (((long)b * 9 + (long)(a1 + da - 1) * 3 + (a2 + db - 1)) << 16);
      const float* kb = kk + (da * 3 + db) * 625;
      for (int d1 = d1lo; d1 < d1hi; ++d1)
        for (int d2 = d2lo; d2 < d2hi; ++d2)
          for (int d3 = d3lo; d3 < d3hi; ++d3) {
            const float* ir = ib + ((((i + d1 - 2) * 16 + (j + d2 - 2)) * 16 + (k + d3 - 2)) << 4) + (l - 2);
            const float* kr = kb + ((d1 * 5 + d2) * 5 + d3) * 5;
            for (int d4 = d4lo; d4 < d4hi; ++d4) sum += ir[d4] * kr[d4];
          }
    }
  sum += bias[0];
  out[idx] = 1.f / (1.f + __expf(-sum));
}

// max over the 9 (scale,scale) channels: [B,9,65536] -> [B,65536]
__global__ void max9_kernel(const float* __restrict__ in, float* __restrict__ out)
{
  int idx = blockIdx.x * blockDim.x + threadIdx.x;
  if (idx >= 4 * 65536) return;
  int b = idx >> 16, sp = idx & 65535;
  const float* p = in + ((long)b * 9 << 16) + sp;
  float m = -1e30f;
  for (int t = 0; t < 9; ++t) m = fmaxf(m, p[(long)t << 16]);
  out[idx] = m;
}

// 4D conv ('same', kernel [5,5,5,5] in LDS) + bias + softplus. in/out [B,32,32,32,32]
__global__ void conv4d_kernel(const float* __restrict__ in, const float* __restrict__ k4,
                              const float* __restrict__ bias, float* __restrict__ out)
{
  __shared__ float kk[625];
  for (int t = threadIdx.x; t < 625; t += blockDim.x) kk[t] = k4[t];
  __syncthreads();
  long idx = (long)blockIdx.x * blockDim.x + threadIdx.x;
  if (idx >= (long)4 * 1048576) return;
  int sp = (int)(idx & 1048575); int b = (int)(idx >> 20);
  int l = sp & 31, k = (sp >> 5) & 31, j = (sp >> 10) & 31, i = (sp >> 15) & 31;
  int d1lo = 2 - i > 0 ? 2 - i : 0, d1hi = 34 - i < 5 ? 34 - i : 5;
  int d2lo = 2 - j > 0 ? 2 - j : 0, d2hi = 34 - j < 5 ? 34 - j : 5;
  int d3lo = 2 - k > 0 ? 2 - k : 0, d3hi = 34 - k < 5 ? 34 - k : 5;
  int d4lo = 2 - l > 0 ? 2 - l : 0, d4hi = 34 - l < 5 ? 34 - l : 5;
  const float* ib = in + ((long)b << 20);
  float sum = 0.f;
  for (int d1 = d1lo; d1 < d1hi; ++d1)
    for (int d2 = d2lo; d2 < d2hi; ++d2)
      for (int d3 = d3lo; d3 < d3hi; ++d3) {
        const float* ir = ib + ((((long)(i + d1 - 2) * 32 + (j + d2 - 2)) * 32 + (k + d3 - 2)) << 5) + (l - 2);
        const float* kr = kk + ((d1 * 5 + d2) * 5 + d3) * 5;
        for (int d4 = d4lo; d4 < d4hi; ++d4) sum += ir[d4] * kr[d4];
      }
  float s = sum + bias[0];
  out[idx] = s > 0.f ? s + log1pf(__expf(-s)) : log1pf(__expf(s));
}

__global__ void rowmax_kernel(const float* __restrict__ c4, float* __restrict__ rmax)
{
  int idx = blockIdx.x * blockDim.x + threadIdx.x;
  if (idx >= 4096) return;
  const float* row = c4 + (long)idx * 1024;
  float m = -1e30f;
  for (int j = 0; j < 1024; ++j) m = fmaxf(m, row[j]);
  rmax[idx] = m;
}

__global__ void colmax_kernel(const float* __restrict__ c4, float* __restrict__ cmax)
{
  int idx = blockIdx.x * blockDim.x + threadIdx.x;
  if (idx >= 4096) return;
  int b = idx >> 10, m = idx & 1023;
  const float* p = c4 + (long)b * 1048576 + m;
  float mx = -1e30f;
  for (int j = 0; j < 1024; ++j) mx = fmaxf(mx, p[(long)j * 1024]);
  cmax[idx] = mx;
}

// mutual-NN filter + softmax(x*20) per row; one 256-thread block per (b, n) row.
__global__ void attn_kernel(const float* __restrict__ c4, const float* __restrict__ rmax,
                            const float* __restrict__ cmax, float* __restrict__ attn)
{
  const int n = blockIdx.x, b = blockIdx.y, tid = threadIdx.x;
  const float* row = c4 + ((long)b * 1024 + n) * 1024;
  float* arow = attn + ((long)b * 1024 + n) * 1024;
  float rm = rmax[b * 1024 + n]; if (rm == 0.f) rm = 1e-30f;
  __shared__ float red[256];
  float tmp[4];
  float lmax = -1e30f;
  for (int t = 0; t < 4; ++t) {
    int m = tid + t * 256;
    float cm = cmax[b * 1024 + m]; if (cm == 0.f) cm = 1e-30f;
    float x = row[m];
    float f = x * (x / rm) * (x / cm) * 20.0f;
    tmp[t] = f;
    lmax = fmaxf(lmax, f);
  }
  red[tid] = lmax; __syncthreads();
  for (int s = 128; s > 0; s >>= 1) { if (tid < s) red[tid] = fmaxf(red[tid], red[tid + s]); __syncthreads(); }
  float mx = red[0]; __syncthreads();
  float lsum = 0.f;
  for (int t = 0; t < 4; ++t) { float e = __expf(tmp[t] - mx); tmp[t] = e; lsum += e; }
  red[tid] = lsum; __syncthreads();
  for (int s = 128; s > 0; s >>= 1) { if (tid < s) red[tid] += red[tid + s]; __syncthreads(); }
  float inv = 1.f / red[0];
  for (int t = 0; t < 4; ++t) arow[tid + t * 256] = tmp[t] * inv;
}

// ---------------------------------------------------------------------------
// out[b,c,m] = sum_n v[b,c,n]*attn[b,m,n] — WMMA f32, M=512,N=1024,K=1024.
// TDM stages 16x128 A/B tiles into LDS (TENSORcnt), fragments read via ds_load.
// grid (32, 64, 4), block 32 (one wave).
// ---------------------------------------------------------------------------
__global__ void av_wmma_kernel(const float* __restrict__ v, const float* __restrict__ attn,
                               float* __restrict__ out)
{
  __shared__ float sA[16 * 128];
  __shared__ float sB[16 * 128];
  const int tm = blockIdx.x, tn = blockIdx.y, b = blockIdx.z;
  const int lane = threadIdx.x;
  const int r  = lane & 15;
  const int kh = (lane >> 4) << 1;
  const float* vrow = v + ((long)b * 512 + tm * 16) * 1024;     // 16 A rows
  const float* arow = attn + ((long)b * 1024 + tn * 16) * 1024; // 16 B rows
  const unsigned long long gA = (unsigned long long)(const void*)vrow;
  const unsigned long long gB = (unsigned long long)(const void*)arow;
  const unsigned ldsA = (unsigned)(unsigned long long)(void*)sA; // LDS offset = addr[31:0]
  const unsigned ldsB = (unsigned)(unsigned long long)(void*)sB;
  v8f acc = {};
  for (int k0 = 0; k0 < 1024; k0 += 128) {
#if defined(__gfx1250__)
    asm volatile("s_wait_dscnt 0" ::: "memory");  // WAR: tile reads done before overwrite
#endif
#if HAVE_TDM
    tdm_load_tile16(ldsA, gA + (unsigned long long)k0 * 4u, 128u, 1024u);
    tdm_load_tile16(ldsB, gB + (unsigned long long)k0 * 4u, 128u, 1024u);
    tdm_wait();
#else
    // cooperative fallback: 16x128 f32 tile = 16x32 float4 rows
    for (int t = lane; t < 16 * 32; t += 32) {
      int row = t >> 5, c4 = t & 31;
      ((float4*)sA)[t] = ((const float4*)(vrow + (long)row * 1024 + k0))[c4];
      ((float4*)sB)[t] = ((const float4*)(arow + (long)row * 1024 + k0))[c4];
    }
#endif
    __syncthreads();   // single-wave WG: lowers to S_NOP + waitcnt fence
    for (int kk = 0; kk < 128; kk += 4) {
      v2f a  = *(const v2f*)&sA[r * 128 + kk + kh];   // ds_load_b64
      v2f bv = *(const v2f*)&sB[r * 128 + kk + kh];
      acc = WMMA_F32(a, bv, acc);
    }
    __syncthreads();
  }
  const int mb  = tm * 16 + ((lane >> 4) << 3);
  const int col = tn * 16 + r;
  for (int j = 0; j < 8; ++j)
    out[((long)b * 512 + mb + j) * 1024 + col] = acc[j];
}

// ---------------------------------------------------------------------------
// Host orchestration
// ---------------------------------------------------------------------------
extern "C" void kernel_launch(void* const* d_in, const int* in_sizes, int n_in,
                              void* d_out, int out_size, void* d_ws, size_t ws_size,
                              hipStream_t stream)
{
  const float* src_feat = (const float*)d_in[0];
  const float* trg_feat = (const float*)d_in[1];
  const float* vmat     = (const float*)d_in[2];
  const float* convw[3] = { (const float*)d_in[3], (const float*)d_in[4], (const float*)d_in[5] };
  const float* k6d = (const float*)d_in[6];
  const float* b6d = (const float*)d_in[7];
  const float* k4d = (const float*)d_in[8];
  const float* b4d = (const float*)d_in[9];
  float* W = (float*)d_ws;
  (void)in_sizes; (void)n_in; (void)out_size; (void)ws_size;

  static const int SZv[3]  = { 11, 16, 23 };
  static const int NSv[3]  = { 121, 256, 529 };
  static const int CUMv[3] = { 0, 121, 377 };

  const long OFF_SRC_I = 0;          // [3][B,512,Ns]           1,855,488
  const long OFF_TRG_I = 1855488;    //                         1,855,488
  const long OFF_SRC_C = 3710976;    // [3][B,128,Ns]             463,872
  const long OFF_TRG_C = 4174848;    //                           463,872
  const long OFF_SNORM = 4638720;    // [3][B,Ns]                   3,624
  const long OFF_TNORM = 4642344;    //                             3,624
  const long OFF_CORR  = 4645968;    // 9 pairs [B,Ns,Nt]       3,283,344
  const long OFF_TMP1  = 7929312;    // [B,529,16,16] max         541,696
  const long OFF_C6D   = 8471008;    // [B,3,3,16^4]            2,359,296
  const long OFF_C6DO  = 10830304;   //                         2,359,296
  const long OFF_CMAX  = 13189600;   // [B,16^4]                  262,144
  const long OFF_TMP2  = 13451744;   // [B,256,32,32]           1,048,576
  const long OFF_C4D   = 14500320;   // [B,32^4]                4,194,304
  const long OFF_C4DO  = 18694624;   //                         4,194,304
  const long OFF_RMAX  = 22888928;   // [B,1024]                    4,096
  const long OFF_CMX2  = 22893024;   //                             4,096
  const long OFF_ATTN  = 22897120;   // [B,1024,1024]           4,194,304

  // 1) feature pyramid (bilinear, align_corners)
  for (int si = 0; si < 3; ++si) {
    int sz = SZv[si], Np = NSv[si];
    long tot = 4L * 512 * Np;
    int blk = (int)((tot + 255) / 256);
    bilerp_kernel<<<blk, 256, 0, stream>>>(src_feat, W + OFF_SRC_I + 4L * 512 * CUMv[si],
        4, 512, 16, 16, sz, sz, 512L * 256, 256L, 16L, 1L, 512L * Np, 0);
    bilerp_kernel<<<blk, 256, 0, stream>>>(trg_feat, W + OFF_TRG_I + 4L * 512 * CUMv[si],
        4, 512, 16, 16, sz, sz, 512L * 256, 256L, 16L, 1L, 512L * Np, 0);
  }
  // 2) 3x3 conv 512->128 via WMMA GEMM
  for (int si = 0; si < 3; ++si) {
    dim3 g(8, (NSv[si] + 15) / 16, 4);
    conv2d_wmma_kernel<<<g, 32, 0, stream>>>(W + OFF_SRC_I + 4L * 512 * CUMv[si], convw[si],
        W + OFF_SRC_C + 4L * 128 * CUMv[si], SZv[si], NSv[si]);
    conv2d_wmma_kernel<<<g, 32, 0, stream>>>(W + OFF_TRG_I + 4L * 512 * CUMv[si], convw[si],
        W + OFF_TRG_C + 4L * 128 * CUMv[si], SZv[si], NSv[si]);
  }
  // 3) norms
  for (int si = 0; si < 3; ++si) {
    int tot = 4 * NSv[si];
    int blk = (tot + 255) / 256;
    norm_kernel<<<blk, 256, 0, stream>>>(W + OFF_SRC_C + 4L * 128 * CUMv[si],
                                         W + OFF_SNORM + 4L * CUMv[si], NSv[si]);
    norm_kernel<<<blk, 256, 0, stream>>>(W + OFF_TRG_C + 4L * 128 * CUMv[si],
                                         W + OFF_TNORM + 4L * CUMv[si], NSv[si]);
  }
  // 4) correlations (WMMA) + interpolate4d -> corr6d (clip >= 0)
  long poff = 0;
  for (int si = 0; si < 3; ++si) {
    for (int ti = 0; ti < 3; ++ti) {
      int Ns = NSv[si], Nt = NSv[ti], ss = SZv[si], ts = SZv[ti];
      float* corr = W + OFF_CORR + poff;
      dim3 g((Ns + 15) / 16, (Nt + 15) / 16, 4);
      corr_wmma_kernel<<<g, 32, 0, stream>>>(W + OFF_SRC_C + 4L * 128 * CUMv[si],
          W + OFF_TRG_C + 4L * 128 * CUMv[ti], W + OFF_SNORM + 4L * CUMv[si],
          W + OFF_TNORM + 4L * CUMv[ti], corr, Ns, Nt);
      // stage 1: [B, ch=Nt, ss, ss] -> [B, Nt, 16, 16]
      long t1 = 4L * Nt * 256;
      bilerp_kernel<<<(int)((t1 + 255) / 256), 256, 0, stream>>>(corr, W + OFF_TMP1,
          4, Nt, ss, ss, 16, 16, (long)Ns * Nt, 1L, (long)ss * Nt, (long)Nt, (long)Nt * 256, 0);
      // stage 2: [B, ch=256, ts, ts] -> corr6d[b, si, ti, :, :, 16, 16], clip
      long t2 = 4L * 256 * 256;
      bilerp_kernel<<<(int)((t2 + 255) / 256), 256, 0, stream>>>(W + OFF_TMP1,
          W + OFF_C6D + (long)(si * 3 + ti) * 65536, 4, 256, ts, ts, 16, 16,
          (long)Nt * 256, 1L, (long)ts * 256, 256L, 9L * 65536, 1);
      poff += 4L * Ns * Nt;
    }
  }
  // 5) conv6d + sigmoid, then max over 9
  conv6d_kernel<<<9216, 256, 0, stream>>>(W + OFF_C6D, k6d, b6d, W + OFF_C6DO);
  max9_kernel<<<1024, 256, 0, stream>>>(W + OFF_C6DO, W + OFF_CMAX);
  // 6) interpolate4d 16^4 -> 32^4
  bilerp_kernel<<<(int)((4L * 256 * 1024 + 255) / 256), 256, 0, stream>>>(W + OFF_CMAX, W + OFF_TMP2,
      4, 256, 16, 16, 32, 32, 65536L, 1L, 4096L, 256L, 256L * 1024, 0);
  bilerp_kernel<<<(int)((4L * 1024 * 1024 + 255) / 256), 256, 0, stream>>>(W + OFF_TMP2, W + OFF_C4D,
      4, 1024, 16, 16, 32, 32, 262144L, 1L, 16384L, 1024L, 1048576L, 0);
  // 7) conv4d + softplus
  conv4d_kernel<<<16384, 256, 0, stream>>>(W + OFF_C4D, k4d, b4d, W + OFF_C4DO);
  // 8) mutual-NN maxes + filtered softmax
  rowmax_kernel<<<16, 256, 0, stream>>>(W + OFF_C4DO, W + OFF_RMAX);
  colmax_kernel<<<16, 256, 0, stream>>>(W + OFF_C4DO, W + OFF_CMX2);
  dim3 ga(1024, 4);
  attn_kernel<<<ga, 256, 0, stream>>>(W + OFF_C4DO, W + OFF_RMAX, W + OFF_CMX2, W + OFF_ATTN);
  // 9) out = v @ attn^T  (WMMA + TDM-staged LDS tiles)
  dim3 gg(32, 64, 4);
  av_wmma_kernel<<<gg, 32, 0, stream>>>(vmat, W + OFF_ATTN, (float*)d_out);
}